// BasicCount_22359599743499
// MI455X (gfx1250) — compile-verified
//
#include <hip/hip_runtime.h>
#include <stdint.h>

// ---------------- problem constants ----------------
#define B_DIM  64
#define N_DIM  16384
#define C_DIM  100

#define TILE_ROWS         64                 // rows per TDM tile
#define BLOCKS_PER_BATCH  32
#define TILES_PER_BLOCK   (N_DIM / (TILE_ROWS * BLOCKS_PER_BATCH))   // = 8
#define TILE_ELEMS        (TILE_ROWS * C_DIM)                         // 6400 dwords
// tile bytes = 25600; double buffer = 51200 B LDS + 400 B histogram

typedef unsigned int u32x4 __attribute__((ext_vector_type(4)));
typedef int          i32x8 __attribute__((ext_vector_type(8)));
typedef int          i32x4 __attribute__((ext_vector_type(4)));

// Build a TDM descriptor (D#) for a contiguous 1-row tile of TILE_ELEMS dwords
// and issue TENSOR_LOAD_TO_LDS.  Layout per CDNA5 ISA ch.8:
//   group0: count=1 | lds_addr | global_addr[31:0] | global_addr[56:32]+type=2
//   group1: data_size=4B; tensor_dim0=TILE_ELEMS; tensor_dim1=1;
//           tile_dim0=TILE_ELEMS; tile_dim1=1; tensor_dim0_stride=TILE_ELEMS
//   group2/3: higher dims = 1 / unused
__device__ __forceinline__ void tdm_load_tile(uint32_t lds_byte_off, const float* gsrc) {
  uint64_t ga = (uint64_t)(uintptr_t)gsrc;
  u32x4 g0;
  g0[0] = 1u;                                               // count=1, user descriptor
  g0[1] = lds_byte_off;                                     // lds_addr (bytes)
  g0[2] = (uint32_t)ga;                                     // global_addr[31:0]
  g0[3] = (uint32_t)((ga >> 32) & 0x01FFFFFFu) | (2u << 30);// global_addr[56:32], type=2

  i32x8 g1;
  g1[0] = (int)(2u << 16);                                  // data_size = 2 (4 bytes)
  g1[1] = (int)(((unsigned)TILE_ELEMS & 0xFFFFu) << 16);    // tensor_dim0[15:0] @ bits63:48
  g1[2] = (int)((((unsigned)TILE_ELEMS >> 16) & 0xFFFFu)    // tensor_dim0[31:16]
                | (1u << 16));                              // tensor_dim1[15:0] = 1
  g1[3] = (int)(((unsigned)TILE_ELEMS & 0xFFFFu) << 16);    // tile_dim0 @ bits127:112
  g1[4] = 1;                                                // tile_dim1 = 1, tile_dim2 = 0
  g1[5] = TILE_ELEMS;                                       // tensor_dim0_stride[31:0]
  g1[6] = 0;                                                // stride0 hi | stride1 lo
  g1[7] = 0;

  i32x4 g2;
  g2[0] = 1;                                                // tensor_dim2 = 1
  g2[1] = 1;                                                // tensor_dim3 = 1
  g2[2] = 0;
  g2[3] = 0;

  i32x4 g3;
  g3[0] = 0;
  g3[1] = (int)(1u << 16);                                  // tensor_dim4 = 1
  g3[2] = 0;
  g3[3] = 0;

  i32x8 g4;                                                 // extra operand group
  g4[0] = 0; g4[1] = 0; g4[2] = 0; g4[3] = 0;               // (clang-23 6-arg form;
  g4[4] = 0; g4[5] = 0; g4[6] = 0; g4[7] = 0;               //  zero-filled, unused)

  __builtin_amdgcn_tensor_load_to_lds(g0, g1, g2, g3, g4, 0);
}

__global__ void __launch_bounds__(64)
argmax_hist_kernel(const float* __restrict__ in, int* __restrict__ counts) {
  __shared__ float tile[2][TILE_ELEMS];
  __shared__ int   shist[C_DIM];

  const int b   = blockIdx.x;   // batch
  const int blk = blockIdx.y;   // block within batch
  const int tid = threadIdx.x;  // 0..63

  for (int i = tid; i < C_DIM; i += 64) shist[i] = 0;

  const float* gbase =
      in + ((size_t)b * N_DIM + (size_t)blk * (TILES_PER_BLOCK * TILE_ROWS)) * C_DIM;

  const uint32_t ldsOff0 = (uint32_t)(uintptr_t)&tile[0][0];
  const uint32_t ldsOff1 = (uint32_t)(uintptr_t)&tile[1][0];

  const bool wave0 = (tid < 32);  // wave-uniform on wave32

  if (wave0) tdm_load_tile(ldsOff0, gbase);   // prefetch tile 0

  for (int k = 0; k < TILES_PER_BLOCK; ++k) {
    if (wave0) {
      if (k + 1 < TILES_PER_BLOCK) {
        // prefetch next tile into the other buffer, then wait so that the
        // older TDM (tile k) is complete: TENSORcnt <= 1
        tdm_load_tile((k & 1) ? ldsOff0 : ldsOff1,
                      gbase + (size_t)(k + 1) * TILE_ELEMS);
        __builtin_amdgcn_s_wait_tensorcnt(1);
      } else {
        __builtin_amdgcn_s_wait_tensorcnt(0);
      }
    }
    __syncthreads();  // publish LDS tile k to all waves

    // each thread argmaxes one 100-float row; row start = tid*400 B (16B aligned)
    const float4* row4 =
        reinterpret_cast<const float4*>(&tile[k & 1][tid * C_DIM]);
    float best = -3.402823466e+38f;
    int   bi   = 0;
#pragma unroll
    for (int q = 0; q < C_DIM / 4; ++q) {
      float4 v = row4[q];
      int c0 = q * 4;
      if (v.x > best) { best = v.x; bi = c0;     }
      if (v.y > best) { best = v.y; bi = c0 + 1; }
      if (v.z > best) { best = v.z; bi = c0 + 2; }
      if (v.w > best) { best = v.w; bi = c0 + 3; }
    }
    atomicAdd(&shist[bi], 1);
    __syncthreads();  // histogram done + safe to overwrite buffer k&1
  }

  for (int i = tid; i < C_DIM; i += 64)
    atomicAdd(&counts[b * C_DIM + i], shist[i]);
}

__global__ void zero_kernel(int* __restrict__ p, int n) {
  int i = blockIdx.x * blockDim.x + threadIdx.x;
  if (i < n) p[i] = 0;
}

__global__ void normalize_kernel(const int* __restrict__ counts,
                                 float* __restrict__ out, int n, float inv) {
  int i = blockIdx.x * blockDim.x + threadIdx.x;
  if (i < n) out[i] = (float)counts[i] * inv;
}

extern "C" void kernel_launch(void* const* d_in, const int* in_sizes, int n_in,
                              void* d_out, int out_size, void* d_ws, size_t ws_size,
                              hipStream_t stream) {
  (void)in_sizes; (void)n_in; (void)out_size; (void)ws_size;

  const float* in  = (const float*)d_in[0];
  float* out       = (float*)d_out;
  int*   counts    = (int*)d_ws;          // 64*100 ints of scratch

  const int HC = B_DIM * C_DIM;           // 6400

  zero_kernel<<<(HC + 255) / 256, 256, 0, stream>>>(counts, HC);

  dim3 grid(B_DIM, BLOCKS_PER_BATCH, 1);
  argmax_hist_kernel<<<grid, 64, 0, stream>>>(in, counts);

  normalize_kernel<<<(HC + 255) / 256, 256, 0, stream>>>(counts, out, HC,
                                                         1.0f / (float)N_DIM);
}